// SCDM_89318139888190
// MI455X (gfx1250) — compile-verified
//
#include <hip/hip_runtime.h>
#include <math.h>

typedef __attribute__((ext_vector_type(2))) float v2f;
typedef __attribute__((ext_vector_type(8))) float v8f;

#define TILE_M 128
#define TILE_N 64
#define KC 32
#define SA_STRIDE 36   // KC + 4 pad: 144B row stride (16B-aligned), bank-shifted rows
#define SB_STRIDE 68   // TILE_N + 4 pad: 272B row stride (16B-aligned)

// C[M,N] = A[M,K] * B[K,N] (+ bias[n] if add_bias)
// grid = (N/64, M/128), block = 256 threads = 8 waves (4x2 grid of 32x32 wave patches).
// Each wave does 2x2 register blocking of 16x16 WMMA tiles: 4 accumulators,
// 2 A-frags + 2 B-frags -> 4 v_wmma per fragment fetch (2 LDS dwords/lane/wmma).
__global__ __launch_bounds__(256)
void gemm_f32_wmma(const float* __restrict__ A, const float* __restrict__ Bm,
                   const float* __restrict__ bias, float* __restrict__ C,
                   int M, int N, int K, int add_bias)
{
    __shared__ float sA[TILE_M * SA_STRIDE];   // 128 x 32 (+pad)
    __shared__ float sB[KC * SB_STRIDE];       //  32 x 64 (+pad)

    const unsigned tid  = threadIdx.x;
    const unsigned lane = tid & 31u;
    const unsigned wid  = tid >> 5;       // 0..7
    const unsigned wy   = wid >> 1;       // 0..3  (32-row patch)
    const unsigned wx   = wid & 1u;       // 0..1  (32-col patch)

    const unsigned n0 = blockIdx.x * TILE_N;
    const unsigned m0 = blockIdx.y * TILE_M;

    v8f acc00 = {}, acc01 = {}, acc10 = {}, acc11 = {};

    const unsigned koff = (lane < 16u) ? 0u : 2u;   // ISA f32 16x16x4 frag layout
    const unsigned l15  = lane & 15u;
    const unsigned arow0 = wy * 32u + l15;          // mi = 0
    const unsigned arow1 = arow0 + 16u;             // mi = 1
    const unsigned bcol0 = wx * 32u + l15;          // ni = 0
    const unsigned bcol1 = bcol0 + 16u;             // ni = 1

    for (unsigned k0 = 0; k0 < (unsigned)K; k0 += KC) {
        // ---- stage A chunk: 128x32 = 1024 float4, 4 per thread ----
        #pragma unroll
        for (unsigned j = 0; j < 4; ++j) {
            unsigned f   = tid + j * 256u;          // 0..1023
            unsigned row = f >> 3;                  // 8 float4 per 32-float row
            unsigned c4  = f & 7u;
            const float4 av = *(const float4*)(A + (size_t)(m0 + row) * K + k0 + c4 * 4u);
            *(float4*)(sA + row * SA_STRIDE + c4 * 4u) = av;
            if (k0 + KC < (unsigned)K && j == 0)    // hint next A chunk -> global_prefetch_b8
                __builtin_prefetch(A + (size_t)(m0 + row) * K + k0 + KC + c4 * 4u, 0, 1);
        }
        // ---- stage B chunk: 32x64 = 512 float4, 2 per thread ----
        #pragma unroll
        for (unsigned j = 0; j < 2; ++j) {
            unsigned f   = tid + j * 256u;          // 0..511
            unsigned row = f >> 4;                  // 16 float4 per 64-float row
            unsigned c4  = f & 15u;
            const float4 bv = *(const float4*)(Bm + (size_t)(k0 + row) * N + n0 + c4 * 4u);
            *(float4*)(sB + row * SB_STRIDE + c4 * 4u) = bv;
        }
        __syncthreads();

        // ---- 8 K-steps of 4, each doing 4 WMMAs on 2 A-frags x 2 B-frags ----
        #pragma unroll
        for (unsigned kk = 0; kk < KC; kk += 4u) {
            v2f a0, a1, b0, b1;
            a0.x = sA[arow0 * SA_STRIDE + kk + koff];
            a0.y = sA[arow0 * SA_STRIDE + kk + koff + 1u];
            a1.x = sA[arow1 * SA_STRIDE + kk + koff];
            a1.y = sA[arow1 * SA_STRIDE + kk + koff + 1u];
            b0.x = sB[(kk + koff)      * SB_STRIDE + bcol0];
            b0.y = sB[(kk + koff + 1u) * SB_STRIDE + bcol0];
            b1.x = sB[(kk + koff)      * SB_STRIDE + bcol1];
            b1.y = sB[(kk + koff + 1u) * SB_STRIDE + bcol1];
            acc00 = __builtin_amdgcn_wmma_f32_16x16x4_f32(false, a0, false, b0, (short)0, acc00, false, false);
            acc01 = __builtin_amdgcn_wmma_f32_16x16x4_f32(false, a0, false, b1, (short)0, acc01, false, false);
            acc10 = __builtin_amdgcn_wmma_f32_16x16x4_f32(false, a1, false, b0, (short)0, acc10, false, false);
            acc11 = __builtin_amdgcn_wmma_f32_16x16x4_f32(false, a1, false, b1, (short)0, acc11, false, false);
        }
        __syncthreads();
    }

    // ---- store 4 tiles; lanes 0-15 -> rows 0..7, lanes 16-31 -> rows 8..15 ----
    const unsigned ccol0 = n0 + wx * 32u + l15;
    const unsigned ccol1 = ccol0 + 16u;
    const unsigned rbase = m0 + wy * 32u + ((lane < 16u) ? 0u : 8u);
    const float bv0 = add_bias ? bias[ccol0] : 0.0f;
    const float bv1 = add_bias ? bias[ccol1] : 0.0f;
    #pragma unroll
    for (int i = 0; i < 8; ++i) {
        C[(size_t)(rbase + i)       * N + ccol0] = acc00[i] + bv0;
        C[(size_t)(rbase + i)       * N + ccol1] = acc01[i] + bv1;
        C[(size_t)(rbase + i + 16u) * N + ccol0] = acc10[i] + bv0;
        C[(size_t)(rbase + i + 16u) * N + ccol1] = acc11[i] + bv1;
    }
}

#define T_LEN 32
#define D_DIM 1024

// One block (256 thr = 8 waves) per (b,f). Fuses: rho -> masked softmax ->
// att-weighted text sum -> two dot products -> tanh scalars -> out.
__global__ __launch_bounds__(256)
void attn_fuse(const float* __restrict__ txt_projB,  // [B,T,D], bias already folded in
               const float* __restrict__ feat_proj,  // [B*F,D]
               const float* __restrict__ textual,    // [B,T,D]
               const int*   __restrict__ q_masks,    // [B,T]
               const float* __restrict__ wvec,       // [D]
               const float* __restrict__ features,   // [B*F,D]
               const float* __restrict__ Wcw, const float* __restrict__ bcw,
               const float* __restrict__ Wcb, const float* __restrict__ bcb,
               float* __restrict__ out, int F)
{
    const unsigned bf   = blockIdx.x;
    const unsigned b    = bf / (unsigned)F;
    const unsigned tid  = threadIdx.x;
    const unsigned lane = tid & 31u;
    const unsigned wid  = tid >> 5;

    __shared__ float s_att[T_LEN];
    __shared__ float s_red[2][8];

    const float* fp  = feat_proj + (size_t)bf * D_DIM;
    const float* tpb = txt_projB + (size_t)b * T_LEN * D_DIM;

    // Phase A: rho[t] = sum_d w[d]*tanh(tp[t,d] + fp[d]); wave w handles t = 4w..4w+3
    #pragma unroll
    for (unsigned i = 0; i < 4; ++i) {
        const unsigned t = wid * 4u + i;
        const float* tp = tpb + (size_t)t * D_DIM;
        float acc = 0.0f;
        #pragma unroll 4
        for (unsigned d = lane; d < D_DIM; d += 32u)
            acc += wvec[d] * tanhf(tp[d] + fp[d]);
        #pragma unroll
        for (int off = 16; off > 0; off >>= 1)
            acc += __shfl_xor(acc, off, 32);
        if (lane == 0) s_att[t] = acc;
    }
    __syncthreads();

    // Phase B: masked softmax over T=32 entirely inside wave 0
    if (wid == 0) {
        float r  = s_att[lane];
        float qm = (float)q_masks[b * T_LEN + lane];
        r += (1.0f - qm) * -1e30f;
        float m = r;
        #pragma unroll
        for (int off = 16; off > 0; off >>= 1)
            m = fmaxf(m, __shfl_xor(m, off, 32));
        float e = __expf(r - m);
        float s = e;
        #pragma unroll
        for (int off = 16; off > 0; off >>= 1)
            s += __shfl_xor(s, off, 32);
        s_att[lane] = e / s;
    }
    __syncthreads();

    // Phase C: txt_h[d] = sum_t att[t]*textual[b,t,d]; fold into the two dot products
    const float* txt = textual + (size_t)b * T_LEN * D_DIM;
    float dotw = 0.0f, dotb = 0.0f;
    #pragma unroll
    for (unsigned j = 0; j < 4; ++j) {
        const unsigned d = tid + j * 256u;
        float acc = 0.0f;
        #pragma unroll
        for (unsigned t = 0; t < T_LEN; ++t)
            acc += s_att[t] * txt[(size_t)t * D_DIM + d];
        dotw += acc * Wcw[d];
        dotb += acc * Wcb[d];
    }
    #pragma unroll
    for (int off = 16; off > 0; off >>= 1) {
        dotw += __shfl_xor(dotw, off, 32);
        dotb += __shfl_xor(dotb, off, 32);
    }
    if (lane == 0) { s_red[0][wid] = dotw; s_red[1][wid] = dotb; }
    __syncthreads();
    if (tid == 0) {
        float a = 0.0f, c = 0.0f;
        #pragma unroll
        for (int i = 0; i < 8; ++i) { a += s_red[0][i]; c += s_red[1][i]; }
        s_red[0][0] = tanhf(a + bcw[0]);
        s_red[1][0] = tanhf(c + bcb[0]);
    }
    __syncthreads();

    const float sw = s_red[0][0];
    const float sb = s_red[1][0];
    const float* fin = features + (size_t)bf * D_DIM;
    float*       o   = out      + (size_t)bf * D_DIM;
    #pragma unroll
    for (unsigned j = 0; j < 4; ++j) {
        const unsigned d = tid + j * 256u;
        o[d] = sw * fin[d] + sb;
    }
}

extern "C" void kernel_launch(void* const* d_in, const int* in_sizes, int n_in,
                              void* d_out, int out_size, void* d_ws, size_t ws_size,
                              hipStream_t stream) {
    (void)in_sizes; (void)n_in; (void)out_size; (void)ws_size;

    const int B = 8, F = 256, T = 32, D = 1024;

    const float* features = (const float*)d_in[0];
    // d_in[1] = masks, unused by the module
    const float* textual  = (const float*)d_in[2];
    const int*   q_masks  = (const int*)  d_in[3];
    const float* Ws       = (const float*)d_in[4];
    const float* W        = (const float*)d_in[5];
    const float* wvec     = (const float*)d_in[6];   // (D,1) -> D floats
    const float* bvec     = (const float*)d_in[7];   // (1,D) -> D floats
    const float* Wcw      = (const float*)d_in[8];
    const float* bcw      = (const float*)d_in[9];
    const float* Wcb      = (const float*)d_in[10];
    const float* bcb      = (const float*)d_in[11];
    float*       outp     = (float*)d_out;

    float* ws        = (float*)d_ws;
    float* txt_projB = ws;                       // B*T*D  = 256K floats (1 MB)
    float* feat_proj = ws + (size_t)B * T * D;   // B*F*D  = 2M floats   (8 MB)

    // txt_projB = textual @ Ws + b   (M = B*T = 256 -> 2 row-blocks)
    gemm_f32_wmma<<<dim3(D / TILE_N, (B * T) / TILE_M), 256, 0, stream>>>(
        textual, Ws, bvec, txt_projB, B * T, D, D, 1);

    // feat_proj = features @ W       (M = B*F = 2048 -> 16 row-blocks)
    gemm_f32_wmma<<<dim3(D / TILE_N, (B * F) / TILE_M), 256, 0, stream>>>(
        features, W, nullptr, feat_proj, B * F, D, D, 0);

    // fused attention + scale/shift  (one block per (b,f))
    attn_fuse<<<dim3(B * F), 256, 0, stream>>>(
        txt_projB, feat_proj, textual, q_masks, wvec,
        features, Wcw, bcw, Wcb, bcb, outp, F);
}